// CombinedLoss_50714973831388
// MI455X (gfx1250) — compile-verified
//
#include <hip/hip_runtime.h>
#include <math.h>

// ---------------------------------------------------------------------------
// Fused L1 + SSIM loss for 64x1x512x512 fp32, MI455X (gfx1250, wave32).
// 11x11 box filters done as banded-matrix matmuls on the f32 WMMA pipe:
//   Out(16x16) = V(16x28) * Q(28x28) * H(28x16), each stage = 7x
//   v_wmma_f32_16x16x4_f32 with f32 accumulate (matches fp32 reference).
// ---------------------------------------------------------------------------

typedef __attribute__((ext_vector_type(2))) float v2f;
typedef __attribute__((ext_vector_type(8))) float v8f;

#define IMG_H   512
#define IMG_W   512
#define N_PIXF  16777216.0f            // 64*512*512
#define N_TILES 65536                  // 64 images * 32*32 tiles
#define WPB     4                      // waves per block
#define NBLOCKS 4096
#define GWAVES  (NBLOCKS * WPB)        // 16384 waves
#define TPW     (N_TILES / GWAVES)     // 4 tiles per wave
#define PST     33                     // LDS row stride in floats (odd: no bank conflicts)
#define PROWS   28
#define PATCH_N (PROWS * PST)          // 924 floats per patch
#define T_N     (16 * PST)             // 528 floats per intermediate tile
#define WAVE_LDS_N 2640                // max(2*PATCH_N, 5*T_N)
#define C1F     1.0e-4f                // 0.01^2
#define C2F     9.0e-4f                // 0.03^2
#define ALPHA_  0.7f

static __device__ __forceinline__ v8f wmma_f32_16x16x4(v2f a, v2f b, v8f c) {
  // (neg_a, A, neg_b, B, c_mod, C, reuse_a, reuse_b)
  return __builtin_amdgcn_wmma_f32_16x16x4_f32(false, a, false, b, (short)0, c,
                                               false, false);
}

extern "C" __global__ __launch_bounds__(WPB * 32)
void ssim_l1_main(const float* __restrict__ pred,
                  const float* __restrict__ targ,
                  float* __restrict__ partial)
{
  __shared__ float smem[WAVE_LDS_N * WPB];
  __shared__ float red[WPB * 2];

  const int tid  = threadIdx.x;
  const int wave = tid >> 5;
  const int lane = tid & 31;
  const int lh   = lane >> 4;   // 0: lanes 0-15, 1: lanes 16-31
  const int l16  = lane & 15;
  float* wl = smem + wave * WAVE_LDS_N;   // private LDS region per wave

  const float inv11 = 1.0f / 11.0f;
  float ssim_acc = 0.0f;
  float l1_acc   = 0.0f;

  const int gwave = blockIdx.x * WPB + wave;

#pragma unroll 1
  for (int it = 0; it < TPW; ++it) {
    const int tile = gwave + GWAVES * it;
    const int img  = tile >> 10;
    const int trc  = tile & 1023;
    const int r0   = (trc >> 5) << 4;     // tile row origin
    const int c0   = (trc & 31) << 4;     // tile col origin
    const long imgBase = (long)img * (IMG_H * IMG_W);

    // ---- Phase A: load 26x26 halo patch (zero padded to 28x33) ------------
    // Also accumulate L1 over the central 16x16 pixels on the fly.
    for (int i = 0; i < (PATCH_N + 31) / 32; ++i) {
      int idx = i * 32 + lane;
      if (idx < PATCH_N) {
        int pr = idx / PST;
        int pc = idx - pr * PST;
        int gr = r0 - 5 + pr;
        int gc = c0 - 5 + pc;
        float xv = 0.0f, yv = 0.0f;
        if (pr < 26 && pc < 26 && gr >= 0 && gr < IMG_H && gc >= 0 && gc < IMG_W) {
          long o = imgBase + (long)gr * IMG_W + gc;
          xv = pred[o];
          yv = targ[o];
        }
        wl[idx]           = xv;   // patch_x
        wl[PATCH_N + idx] = yv;   // patch_y
        if (pr >= 5 && pr < 21 && pc >= 5 && pc < 21)
          l1_acc += fabsf(xv - yv);
      }
    }
    __syncthreads();

    // ---- Phase B: vertical pass  T_q(16x28) = V(16x28) * Q(28x28) ---------
    // A-operand = banded V synthesized from lane id; B-operand = patch data.
    v8f acc[2][5];
#pragma unroll
    for (int n = 0; n < 2; ++n)
#pragma unroll
      for (int q = 0; q < 5; ++q)
        acc[n][q] = (v8f)0.0f;

    const int m = l16;   // A-matrix row for this lane
#pragma unroll
    for (int k = 0; k < 7; ++k) {
      const int p0 = 4 * k + 2 * lh;      // K index of A vgpr0 / B vgpr0
      v2f av;
      av.x = (p0     >= m && p0     <= m + 10) ? inv11 : 0.0f;
      av.y = (p0 + 1 >= m && p0 + 1 <= m + 10) ? inv11 : 0.0f;
#pragma unroll
      for (int n = 0; n < 2; ++n) {
        const int col = 16 * n + l16;
        float x0 = wl[p0 * PST + col];
        float x1 = wl[(p0 + 1) * PST + col];
        float y0 = wl[PATCH_N + p0 * PST + col];
        float y1 = wl[PATCH_N + (p0 + 1) * PST + col];
        v2f bx  = {x0, x1};
        v2f by  = {y0, y1};
        v2f bxx = {x0 * x0, x1 * x1};
        v2f byy = {y0 * y0, y1 * y1};
        v2f bxy = {x0 * y0, x1 * y1};
        acc[n][0] = wmma_f32_16x16x4(av, bx,  acc[n][0]);
        acc[n][1] = wmma_f32_16x16x4(av, by,  acc[n][1]);
        acc[n][2] = wmma_f32_16x16x4(av, bxx, acc[n][2]);
        acc[n][3] = wmma_f32_16x16x4(av, byy, acc[n][3]);
        acc[n][4] = wmma_f32_16x16x4(av, bxy, acc[n][4]);
      }
    }
    __syncthreads();

    // ---- Store T tiles to LDS (aliases the patch region, now dead) --------
#pragma unroll
    for (int q = 0; q < 5; ++q)
#pragma unroll
      for (int n = 0; n < 2; ++n)
#pragma unroll
        for (int e = 0; e < 8; ++e) {
          int mr = e + 8 * lh;
          wl[q * T_N + mr * PST + 16 * n + l16] = acc[n][q][e];
        }
    __syncthreads();

    // ---- Phase C: horizontal pass  Out_q(16x16) = T_q(16x28) * H(28x16) ---
    // A-operand = T data from LDS; B-operand = banded H from lane id.
    v8f outq[5];
#pragma unroll
    for (int q = 0; q < 5; ++q) outq[q] = (v8f)0.0f;

#pragma unroll
    for (int k = 0; k < 7; ++k) {
      const int cc = 4 * k + 2 * lh;
      v2f bh;
      bh.x = (cc     >= l16 && cc     <= l16 + 10) ? inv11 : 0.0f;
      bh.y = (cc + 1 >= l16 && cc + 1 <= l16 + 10) ? inv11 : 0.0f;
#pragma unroll
      for (int q = 0; q < 5; ++q) {
        v2f at;
        at.x = wl[q * T_N + l16 * PST + cc];
        at.y = wl[q * T_N + l16 * PST + cc + 1];
        outq[q] = wmma_f32_16x16x4(at, bh, outq[q]);
      }
    }

    // ---- Phase D: SSIM map per element, accumulate -------------------------
#pragma unroll
    for (int e = 0; e < 8; ++e) {
      float mx = outq[0][e], my = outq[1][e];
      float xx = outq[2][e], yy = outq[3][e], xy = outq[4][e];
      float mx2 = mx * mx, my2 = my * my, mxy = mx * my;
      float num = (2.0f * mxy + C1F) * (2.0f * (xy - mxy) + C2F);
      float den = (mx2 + my2 + C1F) * ((xx - mx2) + (yy - my2) + C2F);
      ssim_acc += num / den;
    }
    __syncthreads();   // before next iteration reuses the LDS region
  }

  // ---- wave32 tree reduction, then per-block partial (no float atomics) ---
  for (int off = 16; off > 0; off >>= 1) {
    ssim_acc += __shfl_down(ssim_acc, off, 32);
    l1_acc   += __shfl_down(l1_acc,   off, 32);
  }
  if (lane == 0) { red[wave * 2] = ssim_acc; red[wave * 2 + 1] = l1_acc; }
  __syncthreads();
  if (tid == 0) {
    float s = 0.0f, l = 0.0f;
    for (int w = 0; w < WPB; ++w) { s += red[w * 2]; l += red[w * 2 + 1]; }
    partial[blockIdx.x * 2]     = s;
    partial[blockIdx.x * 2 + 1] = l;
  }
}

extern "C" __global__ __launch_bounds__(256)
void ssim_l1_finalize(const float* __restrict__ partial, float* __restrict__ out)
{
  __shared__ float ss[256];
  __shared__ float ll[256];
  const int t = threadIdx.x;
  float s = 0.0f, l = 0.0f;
  for (int i = t; i < NBLOCKS; i += 256) {
    s += partial[2 * i];
    l += partial[2 * i + 1];
  }
  ss[t] = s; ll[t] = l;
  __syncthreads();
  for (int off = 128; off > 0; off >>= 1) {
    if (t < off) { ss[t] += ss[t + off]; ll[t] += ll[t + off]; }
    __syncthreads();
  }
  if (t == 0) {
    float ssim_mean = ss[0] / N_PIXF;
    float l1_mean   = ll[0] / N_PIXF;
    out[0] = ALPHA_ * l1_mean + (1.0f - ALPHA_) * (1.0f - ssim_mean);
  }
}

extern "C" void kernel_launch(void* const* d_in, const int* in_sizes, int n_in,
                              void* d_out, int out_size, void* d_ws, size_t ws_size,
                              hipStream_t stream) {
  (void)in_sizes; (void)n_in; (void)out_size; (void)ws_size;
  const float* pred = (const float*)d_in[0];
  const float* targ = (const float*)d_in[1];
  float* partial = (float*)d_ws;   // NBLOCKS*2 floats = 32 KB of scratch
  ssim_l1_main<<<NBLOCKS, WPB * 32, 0, stream>>>(pred, targ, partial);
  ssim_l1_finalize<<<1, 256, 0, stream>>>(partial, (float*)d_out);
}